// WTConv2d_2576980377959
// MI455X (gfx1250) — compile-verified
//
// WTConv2d fused pipeline for MI455X (gfx1250, wave32).
//
// Roofline: 7.8 GFLOP vs >1 GB HBM traffic @ 23.3 TB/s -> memory bound.
// Strategy: fuse haar_fwd into the per-level depthwise conv (coeff patch built
// in LDS, never materialized in HBM), do the 5x5 depthwise conv as a banded
// (Toeplitz) GEMM on V_WMMA_F32_16X16X4_F32 (25 WMMAs per 16x16 tile/plane),
// stage NHWC stores through LDS for 128B-coalesced writes, and use the CDNA5
// Tensor Data Mover (tensor_load_to_lds + s_wait_tensorcnt) to stage the input
// patch of the final (base-conv + inverse-haar + combine) kernel.

#include <hip/hip_runtime.h>
#include <cstdint>

typedef float v2f __attribute__((ext_vector_type(2)));
typedef float v8f __attribute__((ext_vector_type(8)));
typedef unsigned int u32x4 __attribute__((ext_vector_type(4)));
typedef unsigned int u32x8 __attribute__((ext_vector_type(8)));

// ---------------------------------------------------------------------------
// Depthwise 5x5 "SAME" conv on one plane, 16x16 output tile, via WMMA f32
// 16x16x4.  Plane patch lives in LDS: patch[(row)*rowStride + col*xStride],
// row 0 / col 0 correspond to output tile origin minus 2 (halo), zeros where
// the conv would read outside the image (SAME zero padding).
//   out[m,n] = sum_dy sum_{x'=0..19} patch[m+dy][x'] * T_dy[x',n]
//   chunked over K=4: A_c[m,k]=patch[m+dy][4c+k], B_c[k,n]=K[dy][4c+k-n] (banded)
// A frag (f32 16x4): lanes 0-15 hold K=0,1; lanes 16-31 hold K=2,3 (M=lane%16).
// B frag (f32 4x16): lanes 0-15 hold K=0,1; lanes 16-31 hold K=2,3 (N=lane%16).
// D frag: vgpr r, lane L -> M = r + 8*(L>=16), N = L%16.
// ---------------------------------------------------------------------------
__device__ __forceinline__ v8f conv5x5_wmma(const float* __restrict__ patch,
                                            const float* __restrict__ taps, // 25 taps, [dy*5+dx]
                                            int mL, int hi, int rowStride, int xStride) {
  v8f acc = {};
  const int kp = hi * 2;
#pragma unroll
  for (int dy = 0; dy < 5; ++dy) {
    const float* row = patch + (mL + dy) * rowStride;
#pragma unroll
    for (int c = 0; c < 5; ++c) {
      v2f a;
      a.x = row[(4 * c + kp) * xStride];
      a.y = row[(4 * c + kp + 1) * xStride];
      int t0 = 4 * c + kp - mL;          // tap index for B row kp, col n=mL
      int tc0 = min(max(t0, 0), 4);      // clamp so speculative LDS addr is safe
      int tc1 = min(max(t0 + 1, 0), 4);
      float k0 = taps[dy * 5 + tc0];
      float k1 = taps[dy * 5 + tc1];
      v2f b;
      b.x = (t0 >= 0 && t0 <= 4) ? k0 : 0.0f;
      b.y = (t0 >= -1 && t0 <= 3) ? k1 : 0.0f;
      acc = __builtin_amdgcn_wmma_f32_16x16x4_f32(false, a, false, b, (short)0, acc,
                                                  false, false);
    }
  }
  return acc;
}

// ---------------------------------------------------------------------------
// Forward level kernel: haar_fwd (fused, from `prev` = x or previous LL) +
// depthwise 5x5 conv with fused (scale*kernel) taps.
//   prev : (B, 2h, 2w, 128) NHWC
//   cc   : (B, h, w, 512)   conv output, plane index p = c*4 + band
//   llout: (B, h, w, 128)   un-convolved LL (input for next level), may be null
// Block: 256 threads, handles 16x16 coeff tile x 8 channels (= 32 planes).
// ---------------------------------------------------------------------------
#define FCH   8
#define COSTR 22   // LDS row stride of coeff patch (bank-conflict aware)

__global__ __launch_bounds__(256) void wt_fwd_kernel(
    const float* __restrict__ prev, const float* __restrict__ wker,
    const float* __restrict__ wscale, float* __restrict__ cc,
    float* __restrict__ llout, int h, int w, int tilesX) {
  __shared__ float co[32 * 20 * COSTR];   // [plane][20][COSTR]  (56.3 KB)
  __shared__ float kern[32 * 26];         // fused taps per plane (3.3 KB)
  __shared__ float stage[256 * 33];       // [pixel][plane] output staging (33.8 KB)

  const int tile = blockIdx.x;
  const int Y0 = (tile / tilesX) * 16, X0 = (tile % tilesX) * 16;
  const int b = blockIdx.y;
  const int c0 = blockIdx.z * FCH;        // channel group base
  const int tid = threadIdx.x;
  const int hp = 2 * h, wp = 2 * w;

  // fused taps: scale[p] * kernel[t, p]
  for (int i = tid; i < 32 * 25; i += 256) {
    int pl = i / 25, t = i % 25;
    int pg = c0 * 4 + pl;
    kern[pl * 26 + t] = wker[t * 512 + pg] * wscale[pg];
  }
  // haar_fwd into 20x20 coeff patch (zero outside coeff image -> SAME pad)
  for (int i = tid; i < 400 * FCH; i += 256) {
    int ci = i % FCH;
    int pos = i / FCH;
    int py = pos / 20, px = pos % 20;
    int Y = Y0 - 2 + py, X = X0 - 2 + px;
    float av = 0.f, bv = 0.f, cv = 0.f, dv = 0.f;
    if (Y >= 0 && Y < h && X >= 0 && X < w) {
      const float* p = prev + (((size_t)b * hp + 2 * Y) * wp + 2 * X) * 128 + c0 + ci;
      av = p[0];
      bv = p[128];
      cv = p[(size_t)wp * 128];
      dv = p[(size_t)wp * 128 + 128];
    }
    float ll = (av + bv + cv + dv) * 0.5f;
    float lh = (av + bv - cv - dv) * 0.5f;
    float hl = (av - bv + cv - dv) * 0.5f;
    float hh = (av - bv - cv + dv) * 0.5f;
    float* cop = co + (ci * 4) * (20 * COSTR) + py * COSTR + px;
    cop[0 * 20 * COSTR] = ll;
    cop[1 * 20 * COSTR] = lh;
    cop[2 * 20 * COSTR] = hl;
    cop[3 * 20 * COSTR] = hh;
  }
  __syncthreads();

  // conv: each wave owns 4 planes
  const int wv = tid >> 5, lane = tid & 31, mL = lane & 15, hi = lane >> 4;
#pragma unroll
  for (int q = 0; q < 4; ++q) {
    int pl = wv * 4 + q;
    v8f acc = conv5x5_wmma(co + pl * (20 * COSTR), kern + pl * 26, mL, hi, COSTR, 1);
#pragma unroll
    for (int r = 0; r < 8; ++r) {
      int m = r + hi * 8;
      stage[(m * 16 + mL) * 33 + pl] = acc[r];
    }
  }
  __syncthreads();

  // coalesced NHWC store of cc (32 consecutive planes = 128B per pixel)
  const size_t obase = (((size_t)b * h + Y0) * w + X0) * 512 + c0 * 4;
  for (int i = tid; i < 256 * 32; i += 256) {
    int pix = i >> 5, pl = i & 31;
    int py = pix >> 4, px = pix & 15;
    cc[obase + ((size_t)py * w + px) * 512 + pl] = stage[pix * 33 + pl];
  }
  // un-convolved LL plane for the next pyramid level
  if (llout) {
    const size_t lbase = (((size_t)b * h + Y0) * w + X0) * 128 + c0;
    for (int i = tid; i < 256 * FCH; i += 256) {
      int pix = i >> 3, ci = i & 7;
      int py = pix >> 4, px = pix & 15;
      llout[lbase + ((size_t)py * w + px) * 128 + ci] =
          co[(ci * 4) * (20 * COSTR) + (py + 2) * COSTR + (px + 2)];
    }
  }
}

// ---------------------------------------------------------------------------
// Inverse haar: llout(B,2h,2w,128) = haar_inv( cc(B,h,w,512) with ll += nll_in )
// ---------------------------------------------------------------------------
__global__ __launch_bounds__(256) void wt_inv_kernel(
    const float* __restrict__ cc, const float* __restrict__ nll_in,
    float* __restrict__ llout, int h, int w, int total) {
  int idx = blockIdx.x * 256 + threadIdx.x;
  if (idx >= total) return;
  int c = idx & 127;
  int t = idx >> 7;
  int X = t % w; t /= w;
  int Y = t % h;
  int b = t / h;
  size_t cb = ((size_t)b * h + Y) * w + X;
  const float* c4 = cc + cb * 512 + c * 4;   // 4 bands contiguous -> b128 load
  float ll = c4[0];
  if (nll_in) ll += nll_in[cb * 128 + c];
  float lh = c4[1], hl = c4[2], hh = c4[3];
  float av = (ll + lh + hl + hh) * 0.5f;
  float bv = (ll + lh - hl - hh) * 0.5f;
  float cv = (ll - lh + hl - hh) * 0.5f;
  float dv = (ll - lh - hl + hh) * 0.5f;
  int W2 = 2 * w;
  float* o = llout + (((size_t)b * 2 * h + 2 * Y) * W2 + 2 * X) * 128 + c;
  o[0] = av;
  o[128] = bv;
  o[(size_t)W2 * 128] = cv;
  o[(size_t)W2 * 128 + 128] = dv;
}

// ---------------------------------------------------------------------------
// Final kernel: base depthwise conv on x (WMMA, TDM-staged patch) fused with
// the last inverse haar and the output combine:
//   out = base_scale*(conv5x5(x)+bias) + haar_inv(cc0 with ll+=nll0)
// Block: 16x16 output pixels x 16 channels. Interior tiles stage the 20x20x16
// x-patch with TDM tensor_load_to_lds (pad feature -> [y][x][17] LDS layout).
// ---------------------------------------------------------------------------
#define BCH 16
__global__ __launch_bounds__(256) void wt_final_kernel(
    const float* __restrict__ x, const float* __restrict__ bker,
    const float* __restrict__ bbias, const float* __restrict__ bscale,
    const float* __restrict__ cc0, const float* __restrict__ nll0,
    float* __restrict__ out) {
  constexpr int H = 256, W = 256, C = 128, h = 128, w = 128;
  __shared__ __align__(16) float xp[20 * 20 * 17];  // [py][px][ch16 + 1 pad] (27.2 KB)
  __shared__ float kb[16 * 26];
  __shared__ float stage[256 * 17];

  const int tile = blockIdx.x;
  const int Y0 = (tile >> 4) * 16, X0 = (tile & 15) * 16;
  const int b = blockIdx.y;
  const int c0 = blockIdx.z * BCH;
  const int tid = threadIdx.x;

  for (int i = tid; i < 16 * 25; i += 256) {
    int cl = i / 25, t = i % 25;
    kb[cl * 26 + t] = bker[t * 128 + c0 + cl];
  }

  const bool interior = (Y0 >= 2) && (Y0 + 17 < H) && (X0 >= 2) && (X0 + 17 < W);
  if (interior) {
    if (tid < 32) {  // wave 0 issues one TDM op for the whole block patch
      uint32_t lds = (uint32_t)(uintptr_t)(&xp[0]);
      uint64_t ga = (uint64_t)(uintptr_t)(x + (((size_t)b * H + (Y0 - 2)) * W + (X0 - 2)) * C + c0);
      // D# group 0: count=1, lds_addr, global_addr[56:0], type=2
      u32x4 g0;
      g0[0] = 1u;
      g0[1] = lds;
      g0[2] = (uint32_t)ga;
      g0[3] = ((uint32_t)(ga >> 32) & 0x1FFFFFFu) | (2u << 30);
      // D# group 1: data_size=4B, pad_enable, pad_interval=16 dwords,
      // pad_amount=1 dword; tile 16(ch) x 20(x) x 20(y); strides 128, 32768.
      u32x8 g1;
      g1[0] = (2u << 16) | (1u << 20) | (3u << 22);  // 0x00D20000
      g1[1] = 0u;                                    // tensor_dim0 lo16 (=0, dim0=1<<20)
      g1[2] = 16u;                                   // dim0 hi16 | dim1 lo16(=0)
      g1[3] = 16u | (16u << 16);                     // dim1 hi16 | tile_dim0=16
      g1[4] = 20u | (20u << 16);                     // tile_dim1=20 | tile_dim2=20
      g1[5] = 128u;                                  // tensor_dim0_stride lo32
      g1[6] = (32768u & 0xFFFFu) << 16;              // stride0 hi16(0) | stride1 lo16
      g1[7] = 0u;                                    // stride1 hi32
      u32x4 g2;
      g2[0] = (1u << 20);  // tensor_dim2 (large, no OOB clip)
      g2[1] = 0u;
      g2[2] = 0u;
      g2[3] = 0u;
      u32x4 g3;
      g3[0] = 0u; g3[1] = 0u; g3[2] = 0u; g3[3] = 0u;
      asm volatile("tensor_load_to_lds %0, %1, %2, %3"
                   :
                   : "s"(g0), "s"(g1), "s"(g2), "s"(g3)
                   : "memory");
      __builtin_amdgcn_s_wait_tensorcnt(0);
    }
    __syncthreads();
  } else {
    // boundary fallback: bounds-checked loads, zeros = SAME padding
    for (int i = tid; i < 400 * BCH; i += 256) {
      int cl = i & 15, pos = i >> 4;
      int py = pos / 20, px = pos % 20;
      int yy = Y0 - 2 + py, xx = X0 - 2 + px;
      float v = 0.f;
      if (yy >= 0 && yy < H && xx >= 0 && xx < W)
        v = x[(((size_t)b * H + yy) * W + xx) * C + c0 + cl];
      xp[(py * 20 + px) * 17 + cl] = v;
    }
    __syncthreads();
  }

  // base conv: each wave owns 2 channel planes
  const int wv = tid >> 5, lane = tid & 31, mL = lane & 15, hi = lane >> 4;
#pragma unroll
  for (int q = 0; q < 2; ++q) {
    int cl = wv * 2 + q;
    v8f acc = conv5x5_wmma(xp + cl, kb + cl * 26, mL, hi, 20 * 17, 17);
#pragma unroll
    for (int r = 0; r < 8; ++r) {
      int m = r + hi * 8;
      stage[(m * 16 + mL) * 17 + cl] = acc[r];
    }
  }
  __syncthreads();

  // combine: out = scale*(conv+bias) + haar_inv(cc0 with ll+=nll0)
  const size_t obase = (((size_t)b * H + Y0) * W + X0) * C + c0;
  for (int i = tid; i < 256 * BCH; i += 256) {
    int cl = i & 15, pix = i >> 4;
    int py = pix >> 4, px = pix & 15;
    int y = Y0 + py, xx = X0 + px;
    int Yc = y >> 1, Xc = xx >> 1;
    float sy = (y & 1) ? -1.f : 1.f;
    float sx = (xx & 1) ? -1.f : 1.f;
    int c = c0 + cl;
    size_t cb = ((size_t)b * h + Yc) * w + Xc;
    const float* c4 = cc0 + cb * 512 + c * 4;
    float ll = c4[0] + nll0[cb * 128 + c];
    float rec = 0.5f * (ll + sy * c4[1] + sx * c4[2] + (sy * sx) * c4[3]);
    float conv = stage[pix * 17 + cl];
    out[obase + ((size_t)py * W + px) * C + cl] = bscale[c] * (conv + bbias[c]) + rec;
  }
}

// ---------------------------------------------------------------------------
// Host orchestration
// ---------------------------------------------------------------------------
extern "C" void kernel_launch(void* const* d_in, const int* in_sizes, int n_in,
                              void* d_out, int out_size, void* d_ws, size_t ws_size,
                              hipStream_t stream) {
  (void)in_sizes; (void)n_in; (void)out_size; (void)ws_size;
  const float* x     = (const float*)d_in[0];  // (8,256,256,128)
  const float* bker  = (const float*)d_in[1];  // (5,5,1,128)
  const float* bbias = (const float*)d_in[2];  // (128)
  const float* bscal = (const float*)d_in[3];  // (128)
  const float* wk0 = (const float*)d_in[4];    // (5,5,1,512) per level
  const float* wk1 = (const float*)d_in[5];
  const float* wk2 = (const float*)d_in[6];
  const float* ws0 = (const float*)d_in[7];    // (512) per level
  const float* ws1 = (const float*)d_in[8];
  const float* ws2 = (const float*)d_in[9];

  float* ws   = (float*)d_ws;
  float* cc0  = ws;                    // 8*128*128*512 = 67108864
  float* cc1  = cc0 + 67108864;        // 8* 64* 64*512 = 16777216
  float* cc2  = cc1 + 16777216;        // 8* 32* 32*512 =  4194304
  float* ll0  = cc2 + 4194304;         // 8*128*128*128 = 16777216
  float* ll1  = ll0 + 16777216;        // 8* 64* 64*128 =  4194304
  float* nll1 = ll1 + 4194304;         //                  4194304
  float* nll0 = nll1 + 4194304;        //                 16777216
  float* outp = (float*)d_out;

  // forward pyramid: haar + depthwise conv, fused per level
  wt_fwd_kernel<<<dim3(64, 8, 16), 256, 0, stream>>>(x,   wk0, ws0, cc0, ll0,    128, 128, 8);
  wt_fwd_kernel<<<dim3(16, 8, 16), 256, 0, stream>>>(ll0, wk1, ws1, cc1, ll1,     64,  64, 4);
  wt_fwd_kernel<<<dim3( 4, 8, 16), 256, 0, stream>>>(ll1, wk2, ws2, cc2, nullptr, 32,  32, 2);
  // inverse chain
  wt_inv_kernel<<<4096,  256, 0, stream>>>(cc2, nullptr, nll1, 32, 32, 8 * 32 * 32 * 128);
  wt_inv_kernel<<<16384, 256, 0, stream>>>(cc1, nll1,    nll0, 64, 64, 8 * 64 * 64 * 128);
  // base conv + last inverse + combine
  wt_final_kernel<<<dim3(256, 8, 8), 256, 0, stream>>>(x, bker, bbias, bscal, cc0, nll0, outp);
}